// Hbv_18854906429630
// MI455X (gfx1250) — compile-verified
//
#include <hip/hip_runtime.h>
#include <stdint.h>

// ---------------- problem constants (from reference) ----------------
#define NSTEPS   730
#define NGRID    10000
#define LENF     15
#define NEARZERO 1e-05f

// ---------------- tiling ----------------
#define BLOCK        256                     // cells per block (8 wave32)
#define TCHUNK       8                       // timesteps per async chunk
#define FLOATS_PER_T (BLOCK * 3)             // 768 dwords per timestep slab
#define PIECES_PER_T (FLOATS_PER_T / 4)      // 192 x 16B pieces
#define ROW_BYTES    (NGRID * 3 * 4)         // 120000 B per global timestep
#define TOTAL_BYTES  (NSTEPS * ROW_BYTES)    // 87,600,000 B (fits in int)
#define COPIES_PER_THREAD (TCHUNK * PIECES_PER_T / BLOCK)  // 6
#define BUF_FLOATS   (TCHUNK * FLOATS_PER_T) // 6144 floats per buffer

// ---------------- CDNA5 async global->LDS path ----------------
typedef int v4i_t __attribute__((ext_vector_type(4)));
#define AS1 __attribute__((address_space(1)))
#define AS3 __attribute__((address_space(3)))

#if defined(__has_builtin)
#if __has_builtin(__builtin_amdgcn_global_load_async_to_lds_b128)
#define HAVE_ASYNC_LDS 1
#endif
#endif

#ifdef HAVE_ASYNC_LDS
// b128 flavor signature (per clang diagnostic): (v4i AS1*, v4i AS3*, imm, imm)
#define ASYNC_B128(gsrc, ldst)                                            \
  __builtin_amdgcn_global_load_async_to_lds_b128(                         \
      (AS1 v4i_t *)(gsrc), (AS3 v4i_t *)(ldst), 0, 0)
#if __has_builtin(__builtin_amdgcn_s_wait_asynccnt)
#define WAIT_ASYNC(n) __builtin_amdgcn_s_wait_asynccnt(n)
#else
#define WAIT_ASYNC(n) asm volatile("s_wait_asynccnt %0" ::"n"(n) : "memory")
#endif
#else
// Fallback: synchronous vector copy (still correct, no ASYNCcnt needed)
#define ASYNC_B128(gsrc, ldst)                                            \
  do { *(float4 *)(ldst) = *(const float4 *)(gsrc); } while (0)
#define WAIT_ASYNC(n) do { } while (0)
#endif

// Issue one chunk's worth of async 16B copies: x_phy[t0:t0+TCHUNK, g0:g0+256, :]
// Every thread issues exactly COPIES_PER_THREAD ops (uniform ASYNCcnt per wave).
// Out-of-range pieces are clamped to global offset 0; their LDS slots are never
// read (t >= NSTEPS slots are skipped by the compute loop; byte-OOB pieces only
// back cells with g >= NGRID which never compute/store).
__device__ __forceinline__ void issue_chunk(char *xb, float *dst, int t0,
                                            int g0, int tid) {
#pragma unroll
  for (int r = 0; r < COPIES_PER_THREAD; ++r) {
    int j  = r * BLOCK + tid;
    int tl = j / PIECES_PER_T;
    int pc = j - tl * PIECES_PER_T;
    int goff = (t0 + tl) * ROW_BYTES + g0 * 12 + pc * 16;
    bool ok = ((t0 + tl) < NSTEPS) && (goff + 16 <= TOTAL_BYTES);
    char  *gsrc = xb + (ok ? goff : 0);
    float *ldst = dst + (tl * PIECES_PER_T + pc) * 4;
    ASYNC_B128(gsrc, ldst);
  }
}

__global__ __launch_bounds__(BLOCK) void hbv_fused_kernel(
    const float *__restrict__ x_phy,      // (730, 10000, 3)
    const float *__restrict__ phy_static, // (10000, 16)
    float *__restrict__ out) {            // (730, 10000)
  __shared__ float smem[2 * BUF_FLOATS];  // 48 KB double buffer

  const int tid = threadIdx.x;
  const int g0  = blockIdx.x * BLOCK;
  const int g   = g0 + tid;
  const bool active = (g < NGRID);
  char *xb = (char *)x_phy;

  // ---- per-cell parameters (clamped row for inactive lanes) ----
  const float *pr = phy_static + (size_t)(active ? g : (NGRID - 1)) * 16;
  const float BETA   = pr[0]  * 5.0f    + 1.0f;
  const float FC     = pr[1]  * 950.0f  + 50.0f;
  const float K0     = pr[2]  * 0.85f   + 0.05f;
  const float K1     = pr[3]  * 0.49f   + 0.01f;
  const float K2     = pr[4]  * 0.199f  + 0.001f;
  const float LP     = pr[5]  * 0.8f    + 0.2f;
  const float PERC   = pr[6]  * 10.0f;
  const float UZL    = pr[7]  * 100.0f;
  const float TT     = pr[8]  * 5.0f    - 2.5f;
  const float CFMAX  = pr[9]  * 9.5f    + 0.5f;
  const float CFR    = pr[10] * 0.1f;
  const float CWH    = pr[11] * 0.2f;
  const float BETAET = pr[12] * 4.7f    + 0.3f;
  const float Cpar   = pr[13];
  const float aa     = fmaxf(pr[14] * 2.9f, 0.0f) + 0.1f;
  const float theta  = fmaxf(pr[15] * 6.5f, 0.0f) + 0.5f;

  const float invFC   = 1.0f / FC;
  const float invLPFC = 1.0f / (LP * FC);
  const float CFRX    = CFR * CFMAX;

  // ---- gamma routing weights (normalized), kept in VGPRs ----
  float w[LENF];
  {
    const float lg = lgammaf(aa);
    const float lt = __logf(theta);
    float wsum = 0.0f;
#pragma unroll
    for (int k = 0; k < LENF; ++k) {
      const float tg = (float)k + 0.5f;
      const float lw = (aa - 1.0f) * __logf(tg) - tg / theta - lg - aa * lt;
      w[k] = __expf(lw);
      wsum += w[k];
    }
    const float inv = 1.0f / wsum;
#pragma unroll
    for (int k = 0; k < LENF; ++k) w[k] *= inv;
  }

  // ---- state + routing history (registers) ----
  float snow = NEARZERO, melt = NEARZERO, sm = NEARZERO;
  float suz  = NEARZERO, slz  = NEARZERO;
  float hist[LENF];
#pragma unroll
  for (int k = 0; k < LENF; ++k) hist[k] = 0.0f;

  const int nch = (NSTEPS + TCHUNK - 1) / TCHUNK;  // 92

  // prime the pipeline
  issue_chunk(xb, smem, 0, g0, tid);

  for (int c = 0; c < nch; ++c) {
    float *cur = smem + (c & 1) * BUF_FLOATS;
    // overlap next chunk's DMA with this chunk's compute
    if (c + 1 < nch) {
      issue_chunk(xb, smem + ((c + 1) & 1) * BUF_FLOATS, (c + 1) * TCHUNK, g0,
                  tid);
      WAIT_ASYNC(COPIES_PER_THREAD);  // first 6 (chunk c) retired, in order
    } else {
      WAIT_ASYNC(0);
    }
    __syncthreads();  // chunk c visible to all 8 waves

    const int t0 = c * TCHUNK;
#pragma unroll
    for (int tl = 0; tl < TCHUNK; ++tl) {
      const int t = t0 + tl;
      if (t >= NSTEPS) break;  // uniform
      const float p   = cur[tl * FLOATS_PER_T + tid * 3 + 0];
      const float tmn = cur[tl * FLOATS_PER_T + tid * 3 + 1];
      const float pet = cur[tl * FLOATS_PER_T + tid * 3 + 2];

      // ---- HBV step ----
      const float rain     = (tmn >= TT) ? p : 0.0f;
      const float snowfall = p - rain;
      snow += snowfall;
      const float m = fminf(fmaxf(CFMAX * (tmn - TT), 0.0f), snow);
      melt += m;  snow -= m;
      const float refr = fminf(fmaxf(CFRX * (TT - tmn), 0.0f), melt);
      snow += refr;  melt -= refr;
      const float tosoil = fmaxf(melt - CWH * snow, 0.0f);
      melt -= tosoil;
      const float soil_wet = fminf(__expf(BETA * __logf(sm * invFC)), 1.0f);
      const float recharge = (rain + tosoil) * soil_wet;
      sm += rain + tosoil - recharge;
      const float excess = fmaxf(sm - FC, 0.0f);
      sm -= excess;
      const float evapfac = fminf(__expf(BETAET * __logf(sm * invLPFC)), 1.0f);
      const float etact = fminf(sm, pet * evapfac);
      sm = fmaxf(sm - etact, NEARZERO);
      const float capillary =
          fminf(slz, Cpar * slz * (1.0f - fminf(sm * invFC, 1.0f)));
      sm  = fmaxf(sm + capillary, NEARZERO);
      slz = fmaxf(slz - capillary, NEARZERO);
      suz += recharge + excess;
      const float percact = fminf(suz, PERC);
      suz -= percact;
      const float q0 = K0 * fmaxf(suz - UZL, 0.0f);
      suz -= q0;
      const float q1 = K1 * suz;
      suz -= q1;
      slz += percact;
      const float q2 = K2 * slz;
      slz -= q2;
      const float q = q0 + q1 + q2;

      // ---- fused 15-tap gamma routing via register ring ----
#pragma unroll
      for (int k = LENF - 1; k >= 1; --k) hist[k] = hist[k - 1];
      hist[0] = q;
      float qr = 0.0f;
#pragma unroll
      for (int k = 0; k < LENF; ++k) qr = fmaf(w[k], hist[k], qr);

      if (active) out[(size_t)t * NGRID + g] = qr;
    }
    __syncthreads();  // buf (c&1) free for reuse at chunk c+2
  }
}

extern "C" void kernel_launch(void *const *d_in, const int *in_sizes, int n_in,
                              void *d_out, int out_size, void *d_ws,
                              size_t ws_size, hipStream_t stream) {
  const float *x_phy      = (const float *)d_in[0];
  const float *phy_static = (const float *)d_in[1];
  float *out = (float *)d_out;
  (void)in_sizes; (void)n_in; (void)out_size; (void)d_ws; (void)ws_size;

  const int grid = (NGRID + BLOCK - 1) / BLOCK;  // 40 blocks
  hbv_fused_kernel<<<grid, BLOCK, 0, stream>>>(x_phy, phy_static, out);
}